// OnlinePCACell_15539191677238
// MI455X (gfx1250) — compile-verified
//
#include <hip/hip_runtime.h>
#include <math.h>

// Problem constants (match reference)
#define N 8192
#define D 2048
#define K 256
#define NCHUNK 128          // row-chunks for x^T t partial reduction (64 rows each)

typedef __attribute__((ext_vector_type(2))) float v2f;
typedef __attribute__((ext_vector_type(8))) float v8f;

// ---------------- reduction helpers (wave32) ----------------
__device__ __forceinline__ float waveReduce32(float v) {
    #pragma unroll
    for (int o = 16; o > 0; o >>= 1) v += __shfl_xor(v, o, 32);
    return v;
}

__device__ __forceinline__ float blockReduce256(float v) {
    __shared__ float sm[8];
    const int lane = threadIdx.x & 31, wid = threadIdx.x >> 5;
    v = waveReduce32(v);
    if (lane == 0) sm[wid] = v;
    __syncthreads();
    if (wid == 0) {
        v = (lane < 8) ? sm[lane] : 0.0f;
        v = waveReduce32(v);
    }
    return v; // valid on wave 0
}

__device__ __forceinline__ float dot4(float4 p, float4 q) {
    return p.x * q.x + p.y * q.y + p.z * q.z + p.w * q.w;
}

// ---------------- per-column 0.5/||V0[:,k]|| ----------------
__global__ void k_dinv(const float* __restrict__ V0, float* __restrict__ dinv) {
    const int k = blockIdx.x;
    float acc = 0.0f;
    for (int j = threadIdx.x; j < D; j += 256) {
        float w = V0[(size_t)j * K + k];
        acc += w * w;
    }
    acc = blockReduce256(acc);
    if (threadIdx.x == 0) dinv[k] = 0.5f / sqrtf(acc);
}

// vcur[j] = V0[j, kcomp]
__global__ void k_gather(const float* __restrict__ V0, float* __restrict__ vcur, int kcomp) {
    const int j = blockIdx.x * 256 + threadIdx.x;
    vcur[j] = V0[(size_t)j * K + kcomp];
}

// w[m] = Vn[m,:] . vec   (grid = k blocks; b128 loads)
__global__ void k_rowdot(const float* __restrict__ Vn, const float* __restrict__ vec,
                         float* __restrict__ w) {
    const int m = blockIdx.x;
    const float4* row = (const float4*)(Vn + (size_t)m * D);
    const float4* v4  = (const float4*)vec;
    float acc = 0.0f;
    for (int j = threadIdx.x; j < D / 4; j += 256) acc += dot4(row[j], v4[j]);
    acc = blockReduce256(acc);
    if (threadIdx.x == 0) w[m] = acc;
}

// out[i] = x[i,:] . vec   (one wave per row, 8 rows/block, grid = N/8; b128 loads)
__global__ void k_xmatvec(const float* __restrict__ x, const float* __restrict__ vec,
                          float* __restrict__ out) {
    const int wid = threadIdx.x >> 5, lane = threadIdx.x & 31;
    const int i = blockIdx.x * 8 + wid;
    const float4* row = (const float4*)(x + (size_t)i * D);
    const float4* v4  = (const float4*)vec;
    float acc = 0.0f;
    #pragma unroll 4
    for (int j = lane; j < D / 4; j += 32) acc += dot4(row[j], v4[j]);
    acc = waveReduce32(acc);
    if (lane == 0) out[i] = acc;
}

// Dual-output matvec: one sweep of x produces out1 = x@vec1 and out2 = x@vec2.
// Fuses p_k = x@vn_k with t_{k+1} = x@v_{k+1}: saves a full 64MB sweep per iteration.
__global__ void k_xmatvec2(const float* __restrict__ x,
                           const float* __restrict__ vec1, const float* __restrict__ vec2,
                           float* __restrict__ out1, float* __restrict__ out2) {
    const int wid = threadIdx.x >> 5, lane = threadIdx.x & 31;
    const int i = blockIdx.x * 8 + wid;
    const float4* row = (const float4*)(x + (size_t)i * D);
    const float4* v1  = (const float4*)vec1;
    const float4* v2  = (const float4*)vec2;
    float a0 = 0.0f, a1 = 0.0f;
    #pragma unroll 4
    for (int j = lane; j < D / 4; j += 32) {
        float4 xv = row[j];
        a0 += dot4(xv, v1[j]);
        a1 += dot4(xv, v2[j]);
    }
    a0 = waveReduce32(a0);
    a1 = waveReduce32(a1);
    if (lane == 0) { out1[i] = a0; out2[i] = a1; }
}

// dst[i] -= sum_{m<kc} P[m*N+i] * w[m]   (grid = N/256; P columns coalesced)
__global__ void k_sub_Pw(float* __restrict__ dst, const float* __restrict__ P,
                         const float* __restrict__ w, int kc) {
    const int i = blockIdx.x * 256 + threadIdx.x;
    float acc = 0.0f;
    for (int m = 0; m < kc; ++m) acc += P[(size_t)m * N + i] * w[m];
    dst[i] -= acc;
}

// b[m] = P[:,m] . t   (grid = k blocks; b128 loads)
__global__ void k_Pt(const float* __restrict__ P, const float* __restrict__ t,
                     float* __restrict__ bb) {
    const int m = blockIdx.x;
    const float4* col = (const float4*)(P + (size_t)m * N);
    const float4* t4  = (const float4*)t;
    float acc = 0.0f;
    for (int i = threadIdx.x; i < N / 4; i += 256) acc += dot4(col[i], t4[i]);
    acc = blockReduce256(acc);
    if (threadIdx.x == 0) bb[m] = acc;
}

// part[c, j0..j0+3] = sum over row-chunk c of x[i,j]*t[i]  (grid = (D/1024, NCHUNK))
__global__ void k_xTt(const float* __restrict__ x, const float* __restrict__ t,
                      float* __restrict__ part) {
    const int j4 = blockIdx.x * 256 + threadIdx.x;        // float4 column index
    const int i0 = blockIdx.y * (N / NCHUNK);
    float4 acc = {0.f, 0.f, 0.f, 0.f};
    for (int i = i0; i < i0 + N / NCHUNK; ++i) {
        float4 xv = ((const float4*)(x + (size_t)i * D))[j4];
        float ti = t[i];
        acc.x += xv.x * ti; acc.y += xv.y * ti;
        acc.z += xv.z * ti; acc.w += xv.w * ti;
    }
    ((float4*)(part + (size_t)blockIdx.y * D))[j4] = acc;
}

// s[j] = sum_c part[c,j] - sum_{m<kc} Vn[m,j]*b[m]   (grid = D/256)
__global__ void k_sfin(const float* __restrict__ part, const float* __restrict__ Vn,
                       const float* __restrict__ bb, float* __restrict__ s, int kc) {
    const int j = blockIdx.x * 256 + threadIdx.x;
    float acc = 0.0f;
    #pragma unroll 4
    for (int c = 0; c < NCHUNK; ++c) acc += part[(size_t)c * D + j];
    float corr = 0.0f;
    for (int m = 0; m < kc; ++m) corr += Vn[(size_t)m * D + j] * bb[m];
    s[j] = acc - corr;
}

// u = 0.5*v + dinv[k]*s ; Vn[k,:] = u/||u||   (single block, 256 threads)
__global__ void k_updv(const float* __restrict__ vcur, const float* __restrict__ s,
                       const float* __restrict__ dinv, float* __restrict__ Vn, int kc) {
    __shared__ float sm[8];
    __shared__ float rn_s;
    const float di = dinv[kc];
    float u[8];
    float acc = 0.0f;
    #pragma unroll
    for (int r = 0; r < 8; ++r) {
        const int j = threadIdx.x + 256 * r;
        float uu = 0.5f * vcur[j] + di * s[j];
        u[r] = uu;
        acc += uu * uu;
    }
    const int lane = threadIdx.x & 31, wid = threadIdx.x >> 5;
    acc = waveReduce32(acc);
    if (lane == 0) sm[wid] = acc;
    __syncthreads();
    if (threadIdx.x == 0) {
        float tot = 0.0f;
        #pragma unroll
        for (int w = 0; w < 8; ++w) tot += sm[w];
        rn_s = 1.0f / sqrtf(tot);
    }
    __syncthreads();
    const float rn = rn_s;
    #pragma unroll
    for (int r = 0; r < 8; ++r) {
        const int j = threadIdx.x + 256 * r;
        Vn[(size_t)kc * D + j] = u[r] * rn;
    }
}

// out = x @ Vn^T via V_WMMA_F32_16X16X4_F32.
// One 16x16 output tile per wave; 8 waves/block share the A row-tile; grid = 1024.
__global__ void k_project(const float* __restrict__ x, const float* __restrict__ Vn,
                          float* __restrict__ out) {
    const int wg   = blockIdx.x * 8 + (threadIdx.x >> 5);
    const int lane = threadIdx.x & 31;
    const int tileK = wg & (K / 16 - 1);  // 16 tiles along K
    const int tileM = wg >> 4;            // 512 tiles along N
    const int row0 = tileM * 16, col0 = tileK * 16;
    const int ln = lane & 15, hi = lane >> 4;

    // A 16x4 f32 layout: lanes 0-15 -> M=ln, (K=0,K=1); lanes 16-31 -> M=ln, (K=2,K=3)
    const float* ap = x  + (size_t)(row0 + ln) * D + 2 * hi;
    // B 4x16 f32 layout mirrors A with N=ln
    const float* bp = Vn + (size_t)(col0 + ln) * D + 2 * hi;

    v8f acc = {0.f, 0.f, 0.f, 0.f, 0.f, 0.f, 0.f, 0.f};
    #pragma unroll 4
    for (int dd = 0; dd < D; dd += 4) {
        v2f av, bv;
        av.x = ap[dd]; av.y = ap[dd + 1];
        bv.x = bp[dd]; bv.y = bp[dd + 1];
        acc = __builtin_amdgcn_wmma_f32_16x16x4_f32(
            false, av, false, bv, (short)0, acc, false, false);
    }
    // C/D layout: VGPR r -> M = r + 8*hi, N = ln
    #pragma unroll
    for (int r = 0; r < 8; ++r)
        out[(size_t)(row0 + r + 8 * hi) * K + col0 + ln] = acc[r];
}

extern "C" void kernel_launch(void* const* d_in, const int* in_sizes, int n_in,
                              void* d_out, int out_size, void* d_ws, size_t ws_size,
                              hipStream_t stream) {
    (void)in_sizes; (void)n_in; (void)out_size; (void)ws_size;
    const float* x  = (const float*)d_in[0];   // [N, D]
    const float* V0 = (const float*)d_in[1];   // [D, K]
    float* out = (float*)d_out;                // [N, K]

    // workspace layout (floats); ~11.6 MB total, all overwritten before read each call
    float* ws   = (float*)d_ws;
    float* P    = ws;                          // [K][N] deflation vectors p_k (column-contig)
    float* Vn   = P    + (size_t)K * N;        // [K][D] normalized components
    float* t    = Vn   + (size_t)K * D;        // [N]
    float* s    = t    + N;                    // [D]
    float* part = s    + D;                    // [NCHUNK][D]
    float* vcur = part + (size_t)NCHUNK * D;   // [D]
    float* a    = vcur + D;                    // [K]
    float* b    = a    + K;                    // [K]
    float* c    = b    + K;                    // [K]
    float* dinv = c    + K;                    // [K]

    k_dinv<<<K, 256, 0, stream>>>(V0, dinv);

    // prologue: vcur = V0[:,0]; t_raw = x @ vcur
    k_gather<<<D / 256, 256, 0, stream>>>(V0, vcur, 0);
    k_xmatvec<<<N / 8, 256, 0, stream>>>(x, vcur, t);

    for (int k = 0; k < K; ++k) {
        // correct t: t -= P[:,<k] @ (Vn[<k] @ vcur_k)
        if (k) {
            k_rowdot<<<k, 256, 0, stream>>>(Vn, vcur, a);
            k_sub_Pw<<<N / 256, 256, 0, stream>>>(t, P, a, k);
            k_Pt<<<k, 256, 0, stream>>>(P, t, b);
        }
        // s = r^T t = x^T t - Vn^T (P^T t)
        k_xTt<<<dim3(D / 1024, NCHUNK), 256, 0, stream>>>(x, t, part);
        k_sfin<<<D / 256, 256, 0, stream>>>(part, Vn, b, s, k);
        // u = 0.5 v + (0.5/d) s ; Vn[k] = u/||u||
        k_updv<<<1, 256, 0, stream>>>(vcur, s, dinv, Vn, k);
        // c = Vn[<k] @ vn_k
        if (k) k_rowdot<<<k, 256, 0, stream>>>(Vn, Vn + (size_t)k * D, c);
        // fused sweep: P_raw[:,k] = x@vn_k  and  t_raw(next) = x@v_{k+1}
        if (k < K - 1) {
            k_gather<<<D / 256, 256, 0, stream>>>(V0, vcur, k + 1);
            k_xmatvec2<<<N / 8, 256, 0, stream>>>(x, Vn + (size_t)k * D, vcur,
                                                  P + (size_t)k * N, t);
        } else {
            k_xmatvec<<<N / 8, 256, 0, stream>>>(x, Vn + (size_t)k * D, P + (size_t)k * N);
        }
        // correct new deflation column: P[:,k] -= P[:,<k] @ c
        if (k) k_sub_Pw<<<N / 256, 256, 0, stream>>>(P + (size_t)k * N, P, c, k);
    }

    // out = x @ Vn^T  (f32 WMMA, full precision)
    k_project<<<(N / 16) * (K / 16) / 8, 256, 0, stream>>>(x, Vn, out);
}